// MemoryMatrixBlock_37409165148814
// MI455X (gfx1250) — compile-verified
//
#include <hip/hip_runtime.h>

typedef float v2f __attribute__((ext_vector_type(2)));
typedef float v8f __attribute__((ext_vector_type(8)));

#define B_ 16
#define C_ 128
#define H_ 32
#define W_ 32
#define M_ 16
#define S_ 512
#define D_ 8192
#define WSZ 8
#define LAMBDA_ 0.0025f
#define EPS_SHRINK_ 1e-12f
#define EPS_NORM_ 1e-12f

#define CHUNK 256   // K-chunk staged in LDS for kernel 1
#define APAD 2      // LDS row padding (floats) to avoid 64-bank stride conflicts

// ---------------------------------------------------------------------------
// Kernel 1: att partials.  Per window m: C[16x512] = win[16xK] * mem[m]^T.
// grid = (S/64, M, ks); block = 128 (4 waves, 1 N-tile of 16 slots per wave).
// A staged in LDS (shared across the 4 waves); B read as WMMA-layout float2
// from a hoisted per-chunk base pointer so unrolled offsets become immediates.
// Each (kslice, b, m, s) partial is written by exactly one lane: deterministic.
// ---------------------------------------------------------------------------
__global__ __launch_bounds__(128)
void att_partial_kernel(const float* __restrict__ x, const float* __restrict__ mem,
                        float* __restrict__ part, int klen) {
  __shared__ float As[B_][CHUNK + APAD];
  const int t    = threadIdx.x;
  const int wave = t >> 5;
  const int lane = t & 31;
  const int m    = blockIdx.y;
  const int ksl  = blockIdx.z;
  const int n0   = blockIdx.x * 64 + wave * 16;
  const int mh   = m >> 2, mw = m & 3;

  const int rowA = lane & 15;   // A: batch row
  const int col  = lane & 15;   // B: slot column
  const int hi   = lane >> 4;   // lane half selects K pair {0,1} vs {2,3}

  v8f c = {};
  const int    kbase0  = ksl * klen;
  const size_t memBase = (size_t)m * S_ * D_ + (size_t)(n0 + col) * D_;
  const float* ap      = &As[rowA][2 * hi];   // LDS base for A operand

  for (int kc = 0; kc < klen; kc += CHUNK) {
    const int kbase = kbase0 + kc;
    __syncthreads();
    // stage A chunk: win[b][m][kbase .. kbase+CHUNK) gathered from x
    #pragma unroll
    for (int it = 0; it < (B_ * CHUNK) / 128; ++it) {
      int e   = it * 128 + t;
      int row = e >> 8;              // CHUNK == 256
      int kk  = e & (CHUNK - 1);
      int k   = kbase + kk;
      int ch  = k >> 6;              // d = ch*64 + ii*8 + jj
      int rem = k & 63;
      int ii  = rem >> 3;
      int jj  = rem & 7;
      size_t xi = (((size_t)row * C_ + ch) * H_ + (mh * WSZ + ii)) * W_ + (mw * WSZ + jj);
      As[row][kk] = x[xi];
    }
    __syncthreads();

    // hoisted base pointer: unrolled kk*4 byte offsets fold into IOFFSET
    const float* bp = mem + memBase + (size_t)(kbase + 2 * hi);
    #pragma unroll 8
    for (int kk = 0; kk < CHUNK; kk += 4) {
      v2f a = *(const v2f*)(ap + kk);   // ds_load_2addr_b64 (immediate offsets)
      v2f b = *(const v2f*)(bp + kk);   // global_load_b64 (immediate offsets)
      c = __builtin_amdgcn_wmma_f32_16x16x4_f32(false, a, false, b,
                                                (short)0, c, false, false);
    }
  }

  // C layout: VGPR r, lane -> batch row r + 8*hi, slot n0+col
  float* pp = part + (((size_t)ksl * B_ + 8 * hi) * M_ + m) * S_ + (n0 + col);
  #pragma unroll
  for (int r = 0; r < 8; ++r)
    pp[(size_t)r * (M_ * S_)] = c[r];
}

// ---------------------------------------------------------------------------
// Kernel 2: reduce split-K partials, hard-shrink, L1-normalize over slots.
// One wave per (b,m) row; wave32 __shfl_xor reduction. Safe in-place if
// part == att (workspace-too-small fallback).
// ---------------------------------------------------------------------------
__global__ __launch_bounds__(128)
void shrink_norm_kernel(const float* __restrict__ part, float* __restrict__ att, int ks) {
  const int t    = threadIdx.x;
  const int wave = t >> 5;
  const int lane = t & 31;
  const int row  = blockIdx.x * 4 + wave;   // 0..255
  const int b    = row >> 4;
  const int m    = row & 15;

  float v[S_ / 32];
  float asum = 0.f;
  #pragma unroll
  for (int i = 0; i < S_ / 32; ++i) {
    int s = lane + i * 32;
    size_t idx = ((size_t)b * M_ + m) * S_ + s;
    float a = 0.f;
    for (int sl = 0; sl < ks; ++sl)
      a += part[(size_t)sl * (B_ * M_ * S_) + idx];
    float tt  = a - LAMBDA_;
    float val = fmaxf(tt, 0.f) * a / (fabsf(tt) + EPS_SHRINK_);
    v[i]  = val;
    asum += fabsf(val);
  }
  #pragma unroll
  for (int off = 16; off >= 1; off >>= 1)
    asum += __shfl_xor(asum, off, 32);
  float inv = 1.f / fmaxf(asum, EPS_NORM_);
  #pragma unroll
  for (int i = 0; i < S_ / 32; ++i) {
    int s = lane + i * 32;
    att[((size_t)b * M_ + m) * S_ + s] = v[i] * inv;
  }
}

// ---------------------------------------------------------------------------
// Kernel 3: retrieve out = att @ mem[m] fused with window_reverse scatter.
// Per m: GEMM 16(batch) x 8192(D) x 512(S). att for window m staged in LDS.
// grid = (D/64, M); block = 128 (4 waves, 16 d-columns per wave).
// B pointer advances by 4 rows per step; the K-pair partner sits at an
// immediate +D_ (32768B) offset, so each step is two clause-able b32 loads.
// ---------------------------------------------------------------------------
__global__ __launch_bounds__(128)
void retrieve_kernel(const float* __restrict__ att, const float* __restrict__ mem,
                     float* __restrict__ y) {
  __shared__ float Att[B_][S_ + APAD];
  const int t    = threadIdx.x;
  const int wave = t >> 5;
  const int lane = t & 31;
  const int m    = blockIdx.y;
  const int d0   = blockIdx.x * 64 + wave * 16;
  const int mh   = m >> 2, mw = m & 3;
  const int rowA = lane & 15;
  const int col  = lane & 15;
  const int hi   = lane >> 4;

  // stage att[:, m, :] (16 x 512) once
  #pragma unroll
  for (int it = 0; it < (B_ * S_) / 128; ++it) {
    int e   = it * 128 + t;
    int row = e >> 9;              // S_ == 512
    int s   = e & (S_ - 1);
    Att[row][s] = att[((size_t)row * M_ + m) * S_ + s];
  }
  __syncthreads();

  v8f c = {};
  const float* ap = &Att[rowA][2 * hi];
  const float* bp = mem + (size_t)m * S_ * D_ + (size_t)(d0 + col)
                        + (size_t)(2 * hi) * D_;
  #pragma unroll 4
  for (int k = 0; k < S_; k += 4) {
    v2f a = *(const v2f*)(ap + k);
    v2f b;
    b.x = bp[0];        // mem[m][k+2hi  ][d0+col] : 64B coalesced across lanes
    b.y = bp[D_];       // mem[m][k+2hi+1][d0+col] : immediate +32768B
    c = __builtin_amdgcn_wmma_f32_16x16x4_f32(false, a, false, b,
                                              (short)0, c, false, false);
    bp += (size_t)4 * D_;
  }

  // scatter C tile through window_reverse into y
  const int d  = d0 + col;
  const int ch = d >> 6, rem = d & 63, ii = rem >> 3, jj = rem & 7;
  float* yp = y + (((size_t)(8 * hi) * C_ + ch) * H_ + (mh * WSZ + ii)) * W_
                + (mw * WSZ + jj);
  #pragma unroll
  for (int r = 0; r < 8; ++r)
    yp[(size_t)r * (C_ * H_ * W_)] = c[r];
}

// ---------------------------------------------------------------------------
extern "C" void kernel_launch(void* const* d_in, const int* in_sizes, int n_in,
                              void* d_out, int out_size, void* d_ws, size_t ws_size,
                              hipStream_t stream) {
  (void)in_sizes; (void)n_in; (void)out_size;
  const float* x   = (const float*)d_in[0];
  const float* mem = (const float*)d_in[1];
  float* y   = (float*)d_out;
  float* att = y + (size_t)B_ * C_ * H_ * W_;   // outputs: y then att, flat

  const size_t bms = (size_t)B_ * M_ * S_;      // 131072 elements per K-slice
  int ks = 1;
  float* part = att;  // fallback: raw att in output region, normalized in place
  if (ws_size >= bms * sizeof(float)) {
    part = (float*)d_ws;
    while (ks < 8 && (size_t)(ks * 2) * bms * sizeof(float) <= ws_size) ks *= 2;
  }
  const int klen = D_ / ks;                     // multiple of CHUNK for ks in {1,2,4,8}

  dim3 g1(S_ / 64, M_, ks);
  att_partial_kernel<<<g1, 128, 0, stream>>>(x, mem, part, klen);
  shrink_norm_kernel<<<dim3(256 / 4), 128, 0, stream>>>(part, att, ks);
  retrieve_kernel<<<dim3(D_ / 64, M_), 128, 0, stream>>>(att, mem, y);
}